// Encoder_54743653154843
// MI455X (gfx1250) — compile-verified
//
#include <hip/hip_runtime.h>

#define NN 100000
#define NE 1600000
#define ET (NE + NN)          // edges + self loops
#define D1 128
#define D2 64

typedef float v2f __attribute__((ext_vector_type(2)));
typedef float v8f __attribute__((ext_vector_type(8)));

// ---- order-preserving float<->uint mapping for atomic segment-max ----
__device__ __forceinline__ unsigned ford(float x) {
    unsigned u = __float_as_uint(x);
    return (u & 0x80000000u) ? ~u : (u | 0x80000000u);
}
__device__ __forceinline__ float ford_inv(unsigned u) {
    return __uint_as_float((u & 0x80000000u) ? (u & 0x7fffffffu) : ~u);
}

__device__ __forceinline__ void edge_sd(const int* __restrict__ ei, int t, int& s, int& d) {
    if (t < NE) { s = ei[t]; d = ei[NE + t]; }
    else        { s = t - NE; d = t - NE; }      // self loop
}

// ---------------- zero fill ----------------
__global__ void zero_f4(float4* __restrict__ p, int n4) {
    int t = blockIdx.x * blockDim.x + threadIdx.x;
    if (t < n4) p[t] = make_float4(0.f, 0.f, 0.f, 0.f);
}

// Stage W [K x 64] into LDS pre-swizzled into WMMA-B register layout:
// slot p = (kk*4 + nt)*32 + lane holds the v2f this lane feeds to WMMA SRC1
// for k-group kk, n-tile nt:  { W[4kk+koff][col], W[4kk+koff+1][col] },
// koff = 2*(lane>=16), col = nt*16 + (lane&15).  One aligned ds_load_b64,
// 8B lane stride -> all 64 LDS banks hit exactly once (conflict-free).
template<int K>
__device__ __forceinline__ void stage_B(const float* __restrict__ W, float* __restrict__ lB) {
    for (int p = threadIdx.x; p < K * 16; p += 256) {
        int lane = p & 31;
        int nt   = (p >> 5) & 3;
        int kk   = p >> 7;
        int r0   = 4 * kk + ((lane >> 4) << 1);
        int c    = nt * 16 + (lane & 15);
        lB[p * 2]     = W[r0 * 64 + c];
        lB[p * 2 + 1] = W[(r0 + 1) * 64 + c];
    }
}

// ---------------- node transform: H = X @ W  (K x 64), WMMA f32 16x16x4 ----------------
template<int K>
__global__ __launch_bounds__(256) void gemm_node(const float* __restrict__ X,
                                                 const float* __restrict__ W,
                                                 float* __restrict__ H, int nrows) {
    __shared__ float lB[K * 64];
    stage_B<K>(W, lB);
    __syncthreads();

    const int wave = threadIdx.x >> 5;
    const int lane = threadIdx.x & 31;
    const int tile = blockIdx.x * 8 + wave;           // 16 output rows per wave
    if (tile * 16 >= nrows) return;

    const int row  = tile * 16 + (lane & 15);
    const int koff = (lane >> 4) << 1;                // lanes 0-15: K=k,k+1 ; 16-31: K=k+2,k+3
    const int col  = lane & 15;

    v8f acc0 = {}, acc1 = {}, acc2 = {}, acc3 = {};
    const float* xr = X + (long)row * K;
    const float* bp = lB + lane * 2;

    for (int kk = 0; kk < K / 4; kk++) {
        v2f a = *(const v2f*)(xr + 4 * kk + koff);
        v2f b0 = *(const v2f*)(bp + (kk * 4 + 0) * 64);
        v2f b1 = *(const v2f*)(bp + (kk * 4 + 1) * 64);
        v2f b2 = *(const v2f*)(bp + (kk * 4 + 2) * 64);
        v2f b3 = *(const v2f*)(bp + (kk * 4 + 3) * 64);
        acc0 = __builtin_amdgcn_wmma_f32_16x16x4_f32(false, a, false, b0, (short)0, acc0, false, false);
        acc1 = __builtin_amdgcn_wmma_f32_16x16x4_f32(false, a, false, b1, (short)0, acc1, false, false);
        acc2 = __builtin_amdgcn_wmma_f32_16x16x4_f32(false, a, false, b2, (short)0, acc2, false, false);
        acc3 = __builtin_amdgcn_wmma_f32_16x16x4_f32(false, a, false, b3, (short)0, acc3, false, false);
    }
    // C/D layout: VGPR j holds row (j + 8*(lane>=16)), col = lane&15
    const int rbase = tile * 16 + ((lane >> 4) << 3);
    #pragma unroll
    for (int j = 0; j < 8; j++) {
        float* o = H + (long)(rbase + j) * 64 + col;
        o[0] = acc0[j]; o[16] = acc1[j]; o[32] = acc2[j]; o[48] = acc3[j];
    }
}

// ---------------- attention coefficients: a_s = H.att_src, a_d = H.att_dst ----------------
__global__ void attn_coef(const float* __restrict__ H, const float* __restrict__ vs,
                          const float* __restrict__ vd, float* __restrict__ a_s,
                          float* __restrict__ a_d, int n) {
    int t = blockIdx.x * blockDim.x + threadIdx.x;
    if (t >= n) return;
    const float4* h  = (const float4*)(H + (long)t * 64);
    const float4* s4 = (const float4*)vs;
    const float4* d4 = (const float4*)vd;
    float ss = 0.f, dd = 0.f;
    #pragma unroll
    for (int i = 0; i < 16; i++) {
        float4 hv = h[i]; float4 sv = s4[i]; float4 dv = d4[i];
        ss += hv.x * sv.x + hv.y * sv.y + hv.z * sv.z + hv.w * sv.w;
        dd += hv.x * dv.x + hv.y * dv.y + hv.z * dv.z + hv.w * dv.w;
    }
    a_s[t] = ss; a_d[t] = dd;
}

// ---------------- edge pass 1: e = leaky_relu(a_s[s]+a_d[d]); segment max over dst ----------------
__global__ void edge_max(const int* __restrict__ ei, const float* __restrict__ a_s,
                         const float* __restrict__ a_d, float* __restrict__ e_buf,
                         unsigned* __restrict__ m_ord) {
    int t = blockIdx.x * blockDim.x + threadIdx.x;
    if (t >= ET) return;
    int s, d; edge_sd(ei, t, s, d);
    float e = a_s[s] + a_d[d];
    e = (e > 0.f) ? e : 0.2f * e;
    e_buf[t] = e;
    atomicMax(&m_ord[d], ford(e));
}

// ---------------- edge pass 2: ex = exp(e - m[d]); segment sum over dst ----------------
__global__ void edge_expsum(const int* __restrict__ ei, float* __restrict__ e_buf,
                            const unsigned* __restrict__ m_ord, float* __restrict__ denom) {
    int t = blockIdx.x * blockDim.x + threadIdx.x;
    if (t >= ET) return;
    int s, d; edge_sd(ei, t, s, d);
    (void)s;
    float ex = __expf(e_buf[t] - ford_inv(m_ord[d]));
    e_buf[t] = ex;
    atomicAdd(&denom[d], ex);
}

// ---------------- edge pass 3: out[d] += (ex/denom[d]) * h[s]   (16 lanes / edge, float4) ----------------
__global__ void edge_scatter(const int* __restrict__ ei, const float* __restrict__ ex,
                             const float* __restrict__ denom, const float* __restrict__ H,
                             float* __restrict__ OUT) {
    int t = blockIdx.x * blockDim.x + threadIdx.x;
    int e = t >> 4;
    if (e >= ET) return;
    int l = t & 15;
    int s, d; edge_sd(ei, e, s, d);
    float alpha = ex[e] / denom[d];
    float4 hv = *(const float4*)(H + (long)s * 64 + l * 4);
    float* o = OUT + (long)d * 64 + l * 4;
    atomicAdd(o + 0, alpha * hv.x);
    atomicAdd(o + 1, alpha * hv.y);
    atomicAdd(o + 2, alpha * hv.z);
    atomicAdd(o + 3, alpha * hv.w);
}

// ---------------- bias (+ optional relu), in place ----------------
__global__ void bias_act(float* __restrict__ H, const float* __restrict__ B, int n16, int relu) {
    int t = blockIdx.x * blockDim.x + threadIdx.x;
    if (t >= n16) return;
    int n = t >> 4, c = (t & 15) << 2;
    float4* p = (float4*)(H + (long)n * 64 + c);
    float4 bv = *(const float4*)(B + c);
    float4 v = *p;
    v.x += bv.x; v.y += bv.y; v.z += bv.z; v.w += bv.w;
    if (relu) {
        v.x = fmaxf(v.x, 0.f); v.y = fmaxf(v.y, 0.f);
        v.z = fmaxf(v.z, 0.f); v.w = fmaxf(v.w, 0.f);
    }
    *p = v;
}

// ---------------- final: OUT = (OUT + b2) @ Wl + bl, in place, WMMA ----------------
__global__ __launch_bounds__(256) void final_linear(float* __restrict__ OUT,
                                                    const float* __restrict__ B2,
                                                    const float* __restrict__ WL,
                                                    const float* __restrict__ BL, int nrows) {
    __shared__ float lB[64 * 64];
    __shared__ float lb2[64];
    __shared__ float lbl[64];
    stage_B<64>(WL, lB);
    if (threadIdx.x < 64) { lb2[threadIdx.x] = B2[threadIdx.x]; lbl[threadIdx.x] = BL[threadIdx.x]; }
    __syncthreads();

    const int wave = threadIdx.x >> 5;
    const int lane = threadIdx.x & 31;
    const int tile = blockIdx.x * 8 + wave;
    if (tile * 16 >= nrows) return;

    const int row  = tile * 16 + (lane & 15);
    const int koff = (lane >> 4) << 1;
    const int col  = lane & 15;

    // preload all A (with b2 added) before any store: in-place safe within wave
    v2f a[16];
    const float* xr = OUT + (long)row * 64;
    #pragma unroll
    for (int kk = 0; kk < 16; kk++) {
        int kb = 4 * kk + koff;
        v2f v = *(const v2f*)(xr + kb);
        v.x += lb2[kb]; v.y += lb2[kb + 1];
        a[kk] = v;
    }
    v8f acc[4];
    #pragma unroll
    for (int nt = 0; nt < 4; nt++) {
        float b = lbl[nt * 16 + col];
        v8f c = { b, b, b, b, b, b, b, b };
        acc[nt] = c;
    }
    const float* bp = lB + lane * 2;
    #pragma unroll
    for (int kk = 0; kk < 16; kk++) {
        #pragma unroll
        for (int nt = 0; nt < 4; nt++) {
            v2f b = *(const v2f*)(bp + (kk * 4 + nt) * 64);
            acc[nt] = __builtin_amdgcn_wmma_f32_16x16x4_f32(false, a[kk], false, b, (short)0, acc[nt], false, false);
        }
    }
    const int rbase = tile * 16 + ((lane >> 4) << 3);
    #pragma unroll
    for (int j = 0; j < 8; j++) {
        float* o = OUT + (long)(rbase + j) * 64 + col;
        o[0] = acc[0][j]; o[16] = acc[1][j]; o[32] = acc[2][j]; o[48] = acc[3][j];
    }
}

extern "C" void kernel_launch(void* const* d_in, const int* in_sizes, int n_in,
                              void* d_out, int out_size, void* d_ws, size_t ws_size,
                              hipStream_t stream) {
    const float* x   = (const float*)d_in[0];
    const int*   ei  = (const int*)  d_in[1];
    // d_in[2] = edge_attr, unused by reference
    const float* W1  = (const float*)d_in[3];
    const float* as1 = (const float*)d_in[4];
    const float* ad1 = (const float*)d_in[5];
    const float* b1  = (const float*)d_in[6];
    const float* W2  = (const float*)d_in[7];
    const float* as2 = (const float*)d_in[8];
    const float* ad2 = (const float*)d_in[9];
    const float* b2  = (const float*)d_in[10];
    const float* Wl  = (const float*)d_in[11];
    const float* bl  = (const float*)d_in[12];
    float* out = (float*)d_out;

    // workspace carve-up (~60 MB)
    float*    h     = (float*)d_ws;                         // NN*64
    float*    agg   = h + (size_t)NN * 64;                  // NN*64 (layer1 out / layer2 in)
    float*    a_s   = agg + (size_t)NN * 64;                // NN
    float*    a_d   = a_s + NN;                             // NN
    unsigned* m_ord = (unsigned*)(a_d + NN);                // NN
    float*    denom = (float*)(m_ord + NN);                 // NN
    float*    e_buf = denom + NN;                           // ET

    const int TPB = 256;
    const int gTiles  = (NN / 16 + 7) / 8;                  // 782 blocks, 8 waves x 16 rows
    const int gNodes  = (NN + TPB - 1) / TPB;
    const int gEdges  = (ET + TPB - 1) / TPB;
    const int gScat   = ((ET * 16) + TPB - 1) / TPB;
    const int gN16    = (NN * 16 + TPB - 1) / TPB;

    // -------- layer 1 --------
    gemm_node<D1><<<gTiles, TPB, 0, stream>>>(x, W1, h, NN);
    attn_coef<<<gNodes, TPB, 0, stream>>>(h, as1, ad1, a_s, a_d, NN);
    zero_f4<<<(NN * 16 + TPB - 1) / TPB, TPB, 0, stream>>>((float4*)agg, NN * 16);
    zero_f4<<<(NN / 4 + TPB - 1) / TPB, TPB, 0, stream>>>((float4*)m_ord, NN / 4);
    zero_f4<<<(NN / 4 + TPB - 1) / TPB, TPB, 0, stream>>>((float4*)denom, NN / 4);
    edge_max<<<gEdges, TPB, 0, stream>>>(ei, a_s, a_d, e_buf, m_ord);
    edge_expsum<<<gEdges, TPB, 0, stream>>>(ei, e_buf, m_ord, denom);
    edge_scatter<<<gScat, TPB, 0, stream>>>(ei, e_buf, denom, h, agg);
    bias_act<<<gN16, TPB, 0, stream>>>(agg, b1, NN * 16, 1);

    // -------- layer 2 --------
    gemm_node<D2><<<gTiles, TPB, 0, stream>>>(agg, W2, h, NN);
    attn_coef<<<gNodes, TPB, 0, stream>>>(h, as2, ad2, a_s, a_d, NN);
    zero_f4<<<(NN * 16 + TPB - 1) / TPB, TPB, 0, stream>>>((float4*)out, NN * 16);
    zero_f4<<<(NN / 4 + TPB - 1) / TPB, TPB, 0, stream>>>((float4*)m_ord, NN / 4);
    zero_f4<<<(NN / 4 + TPB - 1) / TPB, TPB, 0, stream>>>((float4*)denom, NN / 4);
    edge_max<<<gEdges, TPB, 0, stream>>>(ei, a_s, a_d, e_buf, m_ord);
    edge_expsum<<<gEdges, TPB, 0, stream>>>(ei, e_buf, m_ord, denom);
    edge_scatter<<<gScat, TPB, 0, stream>>>(ei, e_buf, denom, h, out);

    // -------- final linear (adds b2, then @Wl + bl), in place on d_out --------
    final_linear<<<gTiles, TPB, 0, stream>>>(out, b2, Wl, bl, NN);
}